// AntiAliasInterpolation2d_71897752535325
// MI455X (gfx1250) — compile-verified
//
#include <hip/hip_runtime.h>

// (8, 512, 512, 32) f32 -> depthwise 13x13 gaussian blur -> stride-4 subsample
// Fused separable implementation; vertical pass runs on V_WMMA_F32_16X16X4_F32.

typedef __attribute__((ext_vector_type(2))) float v2f;
typedef __attribute__((ext_vector_type(8))) float v8f;

#define CH   32      // channels (== wave32 lanes)
#define HH   512
#define WW   512
#define OH   128
#define OW   128
#define IT   16      // output rows per block   (== WMMA M)
#define JT   8       // output cols per block
#define KCH  19      // K chunks of 4 (covers 73 needed hb rows, padded to 76)
#define RR   (KCH*4) // 76 staged horizontal-blur rows
#define CP   33      // padded channel pitch in LDS (bank-conflict-free)
#define NTHR 256     // 8 waves

__global__ __launch_bounds__(NTHR) void
aa_blur_pool_kernel(const float* __restrict__ x,
                    const float* __restrict__ k2,   // (13,13,1,32) HWIO, same per ch
                    float* __restrict__ out)
{
    extern __shared__ float hb[];   // RR * JT * CP floats = 78.4 KB
    __shared__ float ws[16];        // separable 1-D weights (g / sum(g))

    const int n    = blockIdx.z;
    const int i0   = blockIdx.y * IT;
    const int j0   = blockIdx.x * JT;
    const int tid  = threadIdx.x;
    const int lane = tid & 31;
    const int wv   = tid >> 5;

    // ---- derive separable weights from the provided 2-D kernel -------------
    // k2[u][v] = g[u]g[v]/S with S = (sum g)^2, so column-sum_v = g[v]/sum(g).
    if (tid < 16) {
        float s = 0.f;
        if (tid < 13) {
            #pragma unroll
            for (int u = 0; u < 13; ++u) s += k2[(u * 13 + tid) * CH];  // ch 0
        }
        ws[tid] = s;
    }
    __syncthreads();

    float wr[13];
    #pragma unroll
    for (int t = 0; t < 13; ++t) wr[t] = ws[t];

    // ---- horizontal pass: hb[r][jj][c], 128B coalesced wave loads ----------
    // hb row r <-> absolute input row R = 4*i0 - 6 + r ; sampled col 4*(j0+jj)
    const size_t nbase = (size_t)n * HH * WW * CH;
    for (int rj = wv; rj < RR * JT; rj += NTHR / 32) {
        const int r  = rj >> 3;          // / JT
        const int jj = rj & 7;           // % JT
        const int R  = 4 * i0 - 6 + r;
        float acc = 0.f;
        if (R >= 0 && R < HH) {
            const float* rowp = x + nbase + (size_t)R * WW * CH + lane;
            const int cb = 4 * (j0 + jj) - 6;
            #pragma unroll
            for (int t = 0; t < 13; ++t) {
                const int col = cb + t;                  // zero padding outside
                if (col >= 0 && col < WW)
                    acc = fmaf(wr[t], rowp[(size_t)col * CH], acc);
            }
        }
        hb[(r * JT + jj) * CP + lane] = acc;             // banks 0..31, no conflict
    }
    __syncthreads();

    // ---- vertical pass as banded matmul on V_WMMA_F32_16X16X4_F32 ----------
    // Out(16 out-rows x 16 ch) = G(16 x 76) * HB(76 x 16); G[i][r] = w[r-4i].
    // A 16x4 f32 layout: lanes 0-15 -> M=lane, K = {4k,4k+1}; lanes 16-31 -> {4k+2,4k+3}.
    const int M  = lane & 15;
    const int kh = (lane >> 4) * 2;      // per-lane K sub-offset within a chunk

    float aX[KCH], aY[KCH];              // A fragments: same for every (jj, half)
    #pragma unroll
    for (int k = 0; k < KCH; ++k) {
        const int d0 = 4 * k + kh - 4 * M;
        const int d1 = d0 + 1;
        const int c0i = (d0 < 0) ? 0 : (d0 > 12 ? 12 : d0);
        const int c1i = (d1 < 0) ? 0 : (d1 > 12 ? 12 : d1);
        aX[k] = (d0 >= 0 && d0 <= 12) ? ws[c0i] : 0.f;
        aY[k] = (d1 >= 0 && d1 <= 12) ? ws[c1i] : 0.f;
    }

    // 16 (col, channel-half) tiles per block; 2 per wave; uniform trip count so
    // EXEC is all-ones at every WMMA (ISA requirement).
    for (int p = wv; p < JT * 2; p += NTHR / 32) {
        const int jj = p >> 1;
        const int c0 = (p & 1) << 4;
        const float* bbase = hb + jj * CP + c0 + M;

        v8f acc = {};
        #pragma unroll
        for (int k = 0; k < KCH; ++k) {
            v2f a; a.x = aX[k]; a.y = aY[k];
            const int r0 = 4 * k + kh;
            v2f b;                                   // B 4x16: N = lane&15, K mirrors A
            b.x = bbase[(size_t)(r0    ) * (JT * CP)];
            b.y = bbase[(size_t)(r0 + 1) * (JT * CP)];
            acc = __builtin_amdgcn_wmma_f32_16x16x4_f32(
                      /*neg_a=*/false, a, /*neg_b=*/false, b,
                      /*c_mod=*/(short)0, acc, /*reuse_a=*/false, /*reuse_b=*/false);
        }

        // D layout: VGPR v -> out row i0+v (lanes 0-15) / i0+v+8 (lanes 16-31), N=ch
        const int   ch    = c0 + M;
        const int   mh    = (lane < 16) ? 0 : 8;
        float*      obase = out + (((size_t)n * OH + i0 + mh) * OW + (j0 + jj)) * CH + ch;
        #pragma unroll
        for (int v = 0; v < 8; ++v)
            obase[(size_t)v * OW * CH] = acc[v];
    }
}

extern "C" void kernel_launch(void* const* d_in, const int* in_sizes, int n_in,
                              void* d_out, int out_size, void* d_ws, size_t ws_size,
                              hipStream_t stream) {
    (void)n_in; (void)out_size; (void)d_ws; (void)ws_size;
    const float* x  = (const float*)d_in[0];
    const float* k2 = (const float*)d_in[1];
    float* out = (float*)d_out;

    const int nbatch = in_sizes[0] / (HH * WW * CH);   // 8
    dim3 grid(OW / JT, OH / IT, nbatch);               // (16, 8, 8) = 1024 blocks
    const size_t lds = (size_t)RR * JT * CP * sizeof(float);  // 80,256 B dynamic LDS
    aa_blur_pool_kernel<<<grid, dim3(NTHR), lds, stream>>>(x, k2, out);
}